// RGCNEncoder_50646254354573
// MI455X (gfx1250) — compile-verified
//
#include <hip/hip_runtime.h>

// ---------------------------------------------------------------------------
// RGCN encoder for MI455X (gfx1250, wave32, WMMA).
// All dense 128x128 GEMMs run on V_WMMA_F32_16X16X4_F32.
// ---------------------------------------------------------------------------

constexpr int DIM     = 128;   // feature dim
constexpr int RELS    = 8;     // relations
constexpr int NBASES  = 4;     // bases
constexpr int BT_STR  = 132;   // padded LDS stride for transposed B (bank-conflict free b64)

typedef __attribute__((ext_vector_type(2))) float v2f;
typedef __attribute__((ext_vector_type(8))) float v8f;

// ---------------------------------------------------------------- utilities
__global__ void k_zero(float* __restrict__ p, int n) {
    int i = blockIdx.x * blockDim.x + threadIdx.x;
    if (i < n) p[i] = 0.0f;
}

// W[r] = sum_b comp[r,b] * basis[b]   (R*D*D threads)
__global__ void k_wrel(const float* __restrict__ comp, const float* __restrict__ basis,
                       float* __restrict__ W) {
    int i = blockIdx.x * blockDim.x + threadIdx.x;
    if (i >= RELS * DIM * DIM) return;
    int r  = i / (DIM * DIM);
    int io = i - r * (DIM * DIM);
    float s = 0.0f;
#pragma unroll
    for (int b = 0; b < NBASES; ++b) s += comp[r * NBASES + b] * basis[b * DIM * DIM + io];
    W[i] = s;
}

// deg[dst[e]] += 1
__global__ void k_deg(const int* __restrict__ dst, float* __restrict__ deg, int E) {
    int e = blockIdx.x * blockDim.x + threadIdx.x;
    if (e < E) atomicAdd(&deg[dst[e]], 1.0f);
}

// one wave per edge: agg[dst[e]] += H[src[e]]  if etype[eid[e]] == rel
__global__ void k_scatter(const float* __restrict__ H, const int* __restrict__ src,
                          const int* __restrict__ dst, const int* __restrict__ eid,
                          const int* __restrict__ etype, int rel,
                          float* __restrict__ agg, int E) {
    int t    = blockIdx.x * blockDim.x + threadIdx.x;
    int e    = t >> 5;
    int lane = t & 31;
    if (e >= E) return;
    if (etype[eid[e]] != rel) return;
    const float* hs = H   + (size_t)src[e] * DIM + lane * 4;
    float*       ag = agg + (size_t)dst[e] * DIM + lane * 4;
#pragma unroll
    for (int i = 0; i < 4; ++i) atomicAdd(ag + i, hs[i]);
}

// ------------------------------------------------------------- WMMA GEMM
// C[M x 128] op= A[M x 128] @ B[128 x 128]
//   ADD_BIAS : += bias[col]
//   RELU     : relu at end
//   ACC_SCALE: C = C + (A@B) * (1/max(deg[row],1))   (relation aggregation)
// Block = 256 threads = 8 waves; each wave computes a 16x128 strip.
// B is staged in LDS transposed with padded stride so each B fragment is one
// ds_load_b64; A fragments are one global b64 load per k-step.
template <bool ADD_BIAS, bool RELU, bool ACC_SCALE>
__global__ void k_gemm(const float* __restrict__ A, const float* __restrict__ B,
                       const float* __restrict__ bias, const float* __restrict__ deg,
                       float* __restrict__ C, int M) {
    extern __shared__ float Bt[];   // [128][BT_STR] : Bt[n*BT_STR + k] = B[k][n]
    int tid = threadIdx.x;

    // cooperative transpose-load of B into LDS (coalesced global reads)
#pragma unroll
    for (int i = 0; i < (DIM * DIM) / 256; ++i) {
        int idx = tid + i * 256;
        int k = idx >> 7;
        int n = idx & 127;
        Bt[n * BT_STR + k] = B[idx];
    }
    __syncthreads();

    int wid  = tid >> 5;
    int lane = tid & 31;
    int lcol = lane & 15;       // column / A-row selector within fragment
    int lhi  = lane >> 4;       // upper half-wave flag
    int rowBase = blockIdx.x * 128 + wid * 16;

    // A fragment: lane L holds row (L&15), K-pair 4*kk + 2*(L>>4)  (ISA 16x4 layout)
    int arow = rowBase + lcol;
    if (arow > M - 1) arow = M - 1;           // clamp (stores are guarded)
    const float* Arow = A + (size_t)arow * DIM;

    v8f acc[8];
#pragma unroll
    for (int t = 0; t < 8; ++t)
#pragma unroll
        for (int j = 0; j < 8; ++j) acc[t][j] = 0.0f;

#pragma unroll 4
    for (int kk = 0; kk < DIM / 4; ++kk) {
        int k0 = 4 * kk + 2 * lhi;
        v2f a = *(const v2f*)(Arow + k0);
#pragma unroll
        for (int t = 0; t < 8; ++t) {
            int col = t * 16 + lcol;
            // B fragment: lane L holds col (L&15), K rows k0..k0+1 (ISA 4x16 layout)
            v2f b = *(const v2f*)(Bt + col * BT_STR + k0);
            acc[t] = __builtin_amdgcn_wmma_f32_16x16x4_f32(
                false, a, false, b, (short)0, acc[t], false, false);
        }
    }

    // C layout: VGPR v -> row v + 8*(L>>4), col = 16*t + (L&15)
#pragma unroll
    for (int t = 0; t < 8; ++t) {
        int col = t * 16 + lcol;
        float bv = ADD_BIAS ? bias[col] : 0.0f;
#pragma unroll
        for (int v = 0; v < 8; ++v) {
            int row = rowBase + v + 8 * lhi;
            if (row < M) {
                float val = acc[t][v];
                if (ADD_BIAS) val += bv;
                if (ACC_SCALE) {
                    float s = 1.0f / fmaxf(deg[row], 1.0f);
                    val = C[(size_t)row * DIM + col] + val * s;
                }
                if (RELU) val = fmaxf(val, 0.0f);
                C[(size_t)row * DIM + col] = val;
            }
        }
    }
}

// --------------------------------------------------- LayerNorm + ReLU (wave/row)
__global__ void k_ln_relu(const float* __restrict__ X, const float* __restrict__ g,
                          const float* __restrict__ be, float* __restrict__ Y, int M) {
    int wid  = threadIdx.x >> 5;
    int lane = threadIdx.x & 31;
    int row  = blockIdx.x * 8 + wid;
    if (row >= M) return;
    int c = lane * 4;
    const float* xr = X + (size_t)row * DIM + c;
    float v0 = xr[0], v1 = xr[1], v2 = xr[2], v3 = xr[3];
    float s = v0 + v1 + v2 + v3;
#pragma unroll
    for (int off = 16; off >= 1; off >>= 1) s += __shfl_xor(s, off, 32);
    float mu = s * (1.0f / 128.0f);
    float d0 = v0 - mu, d1 = v1 - mu, d2 = v2 - mu, d3 = v3 - mu;
    float q = d0 * d0 + d1 * d1 + d2 * d2 + d3 * d3;
#pragma unroll
    for (int off = 16; off >= 1; off >>= 1) q += __shfl_xor(q, off, 32);
    float rs = rsqrtf(q * (1.0f / 128.0f) + 1e-5f);
    float* yr = Y + (size_t)row * DIM + c;
    yr[0] = fmaxf(d0 * rs * g[c + 0] + be[c + 0], 0.0f);
    yr[1] = fmaxf(d1 * rs * g[c + 1] + be[c + 1], 0.0f);
    yr[2] = fmaxf(d2 * rs * g[c + 2] + be[c + 2], 0.0f);
    yr[3] = fmaxf(d3 * rs * g[c + 3] + be[c + 3], 0.0f);
}

// --------------------------------------------------------------------- driver
extern "C" void kernel_launch(void* const* d_in, const int* in_sizes, int n_in,
                              void* d_out, int out_size, void* d_ws, size_t ws_size,
                              hipStream_t stream) {
    const float* x     = (const float*)d_in[0];
    const int*   src0  = (const int*)d_in[1];
    const int*   dst0  = (const int*)d_in[2];
    const int*   eid0  = (const int*)d_in[3];
    const int*   src1  = (const int*)d_in[4];
    const int*   dst1  = (const int*)d_in[5];
    const int*   eid1  = (const int*)d_in[6];
    const int*   etype = (const int*)d_in[7];
    const float* preW  = (const float*)d_in[8];
    const float* preB  = (const float*)d_in[9];
    const float* basis0 = (const float*)d_in[10];
    const float* comp0  = (const float*)d_in[11];
    const float* root0  = (const float*)d_in[12];
    const float* bias0  = (const float*)d_in[13];
    const float* g0     = (const float*)d_in[14];
    const float* be0    = (const float*)d_in[15];
    const float* basis1 = (const float*)d_in[16];
    const float* comp1  = (const float*)d_in[17];
    const float* root1  = (const float*)d_in[18];
    const float* bias1  = (const float*)d_in[19];
    const float* g1     = (const float*)d_in[20];
    const float* be1    = (const float*)d_in[21];
    const float* postW  = (const float*)d_in[22];
    const float* postB  = (const float*)d_in[23];
    float* out = (float*)d_out;

    const int N0v = in_sizes[0] / DIM;   // 120000
    const int E0v = in_sizes[1];         // 480000
    const int E1v = in_sizes[4];         // 96000
    const int N1v = 30000, N2v = 6000;

    // workspace layout (floats)
    float* ws = (float*)d_ws;
    size_t off = 0;
    float* h0   = ws + off; off += (size_t)N0v * DIM;   // relu(x@preW+b)
    float* t1   = ws + off; off += (size_t)N1v * DIM;   // layer0 pre-LN accum
    float* h1   = ws + off; off += (size_t)N1v * DIM;   // layer0 output
    float* agg  = ws + off; off += (size_t)N1v * DIM;   // per-relation scatter buf (reused)
    float* deg0 = ws + off; off += (size_t)N1v;
    float* deg1 = ws + off; off += (size_t)N2v;
    float* w0   = ws + off; off += (size_t)RELS * DIM * DIM;
    float* w1   = ws + off; off += (size_t)RELS * DIM * DIM;
    float* t2   = ws + off; off += (size_t)N2v * DIM;   // layer1 pre-LN accum
    float* h2   = ws + off; off += (size_t)N2v * DIM;   // layer1 output

    const size_t ldsB = (size_t)DIM * BT_STR * sizeof(float);
    dim3 B256(256);

    // relation weight composition
    k_wrel<<<dim3((RELS * DIM * DIM + 255) / 256), B256, 0, stream>>>(comp0, basis0, w0);
    k_wrel<<<dim3((RELS * DIM * DIM + 255) / 256), B256, 0, stream>>>(comp1, basis1, w1);

    // degrees
    k_zero<<<dim3((N1v + 255) / 256), B256, 0, stream>>>(deg0, N1v);
    k_zero<<<dim3((N2v + 255) / 256), B256, 0, stream>>>(deg1, N2v);
    k_deg<<<dim3((E0v + 255) / 256), B256, 0, stream>>>(dst0, deg0, E0v);
    k_deg<<<dim3((E1v + 255) / 256), B256, 0, stream>>>(dst1, deg1, E1v);

    // pre: h0 = relu(x @ preW + preB)
    k_gemm<true, true, false><<<dim3((N0v + 127) / 128), B256, ldsB, stream>>>(
        x, preW, preB, nullptr, h0, N0v);

    // ---- layer 0 ----
    k_gemm<true, false, false><<<dim3((N1v + 127) / 128), B256, ldsB, stream>>>(
        h0, root0, bias0, nullptr, t1, N1v);
    for (int r = 0; r < RELS; ++r) {
        k_zero<<<dim3((N1v * DIM + 255) / 256), B256, 0, stream>>>(agg, N1v * DIM);
        k_scatter<<<dim3((E0v * 32 + 255) / 256), B256, 0, stream>>>(
            h0, src0, dst0, eid0, etype, r, agg, E0v);
        k_gemm<false, false, true><<<dim3((N1v + 127) / 128), B256, ldsB, stream>>>(
            agg, w0 + (size_t)r * DIM * DIM, nullptr, deg0, t1, N1v);
    }
    k_ln_relu<<<dim3((N1v + 7) / 8), B256, 0, stream>>>(t1, g0, be0, h1, N1v);

    // ---- layer 1 ----
    k_gemm<true, false, false><<<dim3((N2v + 127) / 128), B256, ldsB, stream>>>(
        h1, root1, bias1, nullptr, t2, N2v);
    for (int r = 0; r < RELS; ++r) {
        k_zero<<<dim3((N2v * DIM + 255) / 256), B256, 0, stream>>>(agg, N2v * DIM);
        k_scatter<<<dim3((E1v * 32 + 255) / 256), B256, 0, stream>>>(
            h1, src1, dst1, eid1, etype, r, agg, E1v);
        k_gemm<false, false, true><<<dim3((N2v + 127) / 128), B256, ldsB, stream>>>(
            agg, w1 + (size_t)r * DIM * DIM, nullptr, deg1, t2, N2v);
    }
    k_ln_relu<<<dim3((N2v + 7) / 8), B256, 0, stream>>>(t2, g1, be1, h2, N2v);

    // post: out = h2 @ postW + postB
    k_gemm<true, false, false><<<dim3((N2v + 127) / 128), B256, ldsB, stream>>>(
        h2, postW, postB, nullptr, out, N2v);

    (void)n_in; (void)out_size; (void)ws_size;
}